// GCNLayer_1236950581457
// MI455X (gfx1250) — compile-verified
//
#include <hip/hip_runtime.h>
#include <cstdint>

// ---------------------------------------------------------------------------
// COO SpMM: out[row[e]] += vals[e] * embeds[col[e]],  D = 32 (== wave32 width)
//
// MI455X / gfx1250 design:
//  - Edge triples streamed via async global->LDS (global_load_async_to_lds_b32,
//    ASYNCcnt), double-buffered per wave, synced with s_wait_asynccnt.
//  - Per 16-edge group: Y = diag(v) * X(gathered 16x32) computed on the matrix
//    pipe as 4 accumulating V_WMMA_F32_16X16X4_F32 per 16x16 tile (rank-16
//    decomposition over K=4), co-executing with VALU address math.
//  - Gathers: 64B-coalesced fp32 row segments (embeds table is L2-resident).
//  - Scatter: non-returning global_atomic_add_f32, resolved in L2.
// ---------------------------------------------------------------------------

#define BLOCK 256
#define WAVES_PER_BLOCK (BLOCK / 32)
#define SPMM_BLOCKS 1024
#define ZERO_BLOCKS 512

typedef float v2f __attribute__((ext_vector_type(2)));
typedef float v8f __attribute__((ext_vector_type(8)));

#if defined(__has_builtin)
#if __has_builtin(__builtin_amdgcn_wmma_f32_16x16x4_f32)
#define HAVE_WMMA_F32X4 1
#endif
#endif

// Per-lane async copy of one dword: copies 4 bytes from the global address in gaddr to the LDS address in lds_off
__device__ __forceinline__ void async_ld_b32(uint32_t lds_off, const uint32_t* gaddr) {
    asm volatile("global_load_async_to_lds_b32 %0, %1, off"
                 :: "v"(lds_off), "v"(gaddr)
                 : "memory");
}
__device__ __forceinline__ void wait_async_3() {
    asm volatile("s_wait_asynccnt 0x3" ::: "memory");
}
__device__ __forceinline__ void wait_async_0() {
    asm volatile("s_wait_asynccnt 0x0" ::: "memory");
}

// Stage one 32-edge batch: rows -> sb[0..31], cols -> sb[32..63], vals -> sb[64..95]
__device__ __forceinline__ void issue_batch(uint32_t* sb,
                                            const uint32_t* rows,
                                            const uint32_t* cols,
                                            const uint32_t* vals,
                                            int64_t base, int lane) {
    // Low 32 bits of a flat shared-memory address == raw LDS byte offset
    // (LDS aperture lives in flat addr[63:32] on CDNA5).
    uint32_t off = (uint32_t)(uintptr_t)(sb + lane);
    async_ld_b32(off,          rows + base + lane);
    async_ld_b32(off + 32 * 4, cols + base + lane);
    async_ld_b32(off + 64 * 4, vals + base + lane);
}

__global__ void __launch_bounds__(BLOCK)
spmm_coo_kernel(const int* __restrict__ adj_row,
                const int* __restrict__ adj_col,
                const float* __restrict__ adj_vals,
                const float* __restrict__ embeds,
                float* __restrict__ out,
                int nEdges) {
    __shared__ uint32_t sbuf[WAVES_PER_BLOCK][2][96];

    const int lane = threadIdx.x & 31;
    const int m    = lane & 15;   // N index within a 16-wide tile / edge sub-index
    const int half = lane >> 4;   // 0: lanes 0-15, 1: lanes 16-31
    const int wv   = threadIdx.x >> 5;
    const int64_t gw = (int64_t)blockIdx.x * WAVES_PER_BLOCK + wv;  // global wave id
    const int64_t nW = (int64_t)gridDim.x * WAVES_PER_BLOCK;
    const int64_t nFull = nEdges / 32;

    const uint32_t* rows = (const uint32_t*)adj_row;
    const uint32_t* cols = (const uint32_t*)adj_col;
    const uint32_t* vals = (const uint32_t*)adj_vals;

    if (gw < nFull) {
        issue_batch(&sbuf[wv][0][0], rows, cols, vals, gw * 32, lane);

        int64_t b = gw;
        int cur = 0;
        for (;;) {
            const int64_t nxt = b + nW;            // wave-uniform
            const bool hasNext = (nxt < nFull);    // wave-uniform
            if (hasNext) {
                issue_batch(&sbuf[wv][cur ^ 1][0], rows, cols, vals, nxt * 32, lane);
                wait_async_3();  // 6 outstanding -> <=3 guarantees batch b landed
            } else {
                wait_async_0();
            }

            const uint32_t* sb = &sbuf[wv][cur][0];

#pragma unroll
            for (int q = 0; q < 32; q += 16) {   // two 16-edge groups per batch
#if HAVE_WMMA_F32X4
                // v_m for this lane's edge sub-index
                const float v = __uint_as_float(sb[64 + q + m]);
                v8f acc0 = {0.f, 0.f, 0.f, 0.f, 0.f, 0.f, 0.f, 0.f}; // feats 0-15
                v8f acc1 = {0.f, 0.f, 0.f, 0.f, 0.f, 0.f, 0.f, 0.f}; // feats 16-31
#pragma unroll
                for (int g = 0; g < 4; ++g) {
                    // B rows 2h,2h+1 of WMMA #g = gathered rows of edges e0,e0+1
                    const int e0 = q + 4 * g + 2 * half;
                    const uint32_t c0 = sb[32 + e0];
                    const uint32_t c1 = sb[32 + e0 + 1];
                    // A[m][k] = v_m iff m == 4g+k  (A.x: K=2h, A.y: K=2h+1)
                    v2f A;
                    A.x = (m == 4 * g + 2 * half)     ? v : 0.f;
                    A.y = (m == 4 * g + 2 * half + 1) ? v : 0.f;
                    const float* p0 = embeds + (size_t)c0 * 32 + m;
                    const float* p1 = embeds + (size_t)c1 * 32 + m;
                    v2f B0, B1;
                    B0.x = p0[0];  B0.y = p1[0];    // tile t=0 (feats 0-15)
                    B1.x = p0[16]; B1.y = p1[16];   // tile t=1 (feats 16-31)
                    acc0 = __builtin_amdgcn_wmma_f32_16x16x4_f32(
                        false, A, false, B0, (short)0, acc0, false, false);
                    acc1 = __builtin_amdgcn_wmma_f32_16x16x4_f32(
                        false, A, false, B1, (short)0, acc1, false, false);
                }
                // D VGPR k, lane -> (M = k+8*half, N = m)
#pragma unroll
                for (int k = 0; k < 8; ++k) {
                    const uint32_t r = sb[q + k + 8 * half];
                    float* po = out + (size_t)r * 32 + m;
                    atomicAdd(po,      acc0[k]);
                    atomicAdd(po + 16, acc1[k]);
                }
#else
                // Fallback: lane = feature, one edge per step
#pragma unroll 8
                for (int j = 0; j < 16; ++j) {
                    const uint32_t r = sb[q + j];
                    const uint32_t c = sb[32 + q + j];
                    const float    vv = __uint_as_float(sb[64 + q + j]);
                    const float    x  = embeds[(size_t)c * 32 + lane];
                    atomicAdd(&out[(size_t)r * 32 + lane], vv * x);
                }
#endif
            }

            if (!hasNext) break;
            b = nxt;
            cur ^= 1;
        }
    }

    // Generic tail (nEdges % 32 != 0); empty for E = 3.2M.
    const int64_t tailStart = nFull * 32;
    if (gw == 0) {
        for (int64_t e = tailStart; e < nEdges; ++e) {
            const uint32_t r = rows[e];
            const uint32_t c = cols[e];
            const float    v = __uint_as_float(vals[e]);
            const float    x = embeds[(size_t)c * 32 + lane];
            atomicAdd(&out[(size_t)r * 32 + lane], v * x);
        }
    }
}

__global__ void __launch_bounds__(BLOCK)
zero_kernel(float* __restrict__ p, int n) {
    const int n4 = n >> 2;
    float4* p4 = (float4*)p;
    int64_t i = (int64_t)blockIdx.x * blockDim.x + threadIdx.x;
    const int64_t stride = (int64_t)gridDim.x * blockDim.x;
    for (; i < n4; i += stride)
        p4[i] = make_float4(0.f, 0.f, 0.f, 0.f);
    int64_t r = (int64_t)(n4 << 2) + (int64_t)blockIdx.x * blockDim.x + threadIdx.x;
    for (; r < n; r += stride)
        p[r] = 0.f;
}

extern "C" void kernel_launch(void* const* d_in, const int* in_sizes, int n_in,
                              void* d_out, int out_size, void* d_ws, size_t ws_size,
                              hipStream_t stream) {
    const int*   adj_row  = (const int*)d_in[0];
    const int*   adj_col  = (const int*)d_in[1];
    const float* adj_vals = (const float*)d_in[2];
    const float* embeds   = (const float*)d_in[3];
    float*       out      = (float*)d_out;
    const int    nEdges   = in_sizes[0];

    zero_kernel<<<ZERO_BLOCKS, BLOCK, 0, stream>>>(out, out_size);
    spmm_coo_kernel<<<SPMM_BLOCKS, BLOCK, 0, stream>>>(adj_row, adj_col, adj_vals,
                                                       embeds, out, nEdges);
}